// LTIModel_59090160058691
// MI455X (gfx1250) — compile-verified
//
#include <hip/hip_runtime.h>
#include <hip/hip_bf16.h>

typedef float v2f __attribute__((ext_vector_type(2)));
typedef float v8f __attribute__((ext_vector_type(8)));

static constexpr int Bn = 128, Tn = 2048, OBSn = 128, Sn = 128, OUTn = 64, Hn = 64;
static constexpr long long ROWS = (long long)Bn * Tn;      // 262144
static constexpr long long OUT_ELEMS = ROWS * 128;         // outputs flat size

#define WMMA_F32X4(a_, b_, c_) \
  __builtin_amdgcn_wmma_f32_16x16x4_f32(false, (a_), false, (b_), (short)0, (c_), false, false)

#define V8F_ZERO {0.f, 0.f, 0.f, 0.f, 0.f, 0.f, 0.f, 0.f}

// ---------------------------------------------------------------------------
// One-time precompute of the exact one-step transition matrices:
//   P = I + hA + h^2/2 A^2 + h^3/6 A^3 + h^4/24 A^4   (A == stored A_T, right-mult)
//   Q = hB + h^2/2 BA + h^3/6 BA^2 + h^4/24 BA^3      (B == stored By_T)
// ---------------------------------------------------------------------------
__global__ void __launch_bounds__(256)
gemm128_acc(float* __restrict__ C, const float* __restrict__ A,
            const float* __restrict__ Bm, const float* __restrict__ dtp,
            int alpha_code, int accumulate)
{
  const int idx = blockIdx.x * 256 + threadIdx.x;    // 0..16383
  const int i = idx >> 7, j = idx & 127;
  const float h = *dtp;
  float alpha = 1.0f;
  if (alpha_code == 1)      alpha = 0.5f * h * h;
  else if (alpha_code == 2) alpha = h * h * h * (1.0f / 6.0f);
  else if (alpha_code == 3) alpha = h * h * h * h * (1.0f / 24.0f);
  float s = 0.0f;
  for (int k = 0; k < 128; ++k) s = fmaf(A[i * 128 + k], Bm[k * 128 + j], s);
  C[idx] = (accumulate ? C[idx] : 0.0f) + alpha * s;
}

__global__ void __launch_bounds__(256)
build_P(float* __restrict__ P, const float* __restrict__ A,
        const float* __restrict__ T2, const float* __restrict__ T3,
        const float* __restrict__ dtp)
{
  const int idx = blockIdx.x * 256 + threadIdx.x;
  const int i = idx >> 7, j = idx & 127;
  const float h = *dtp;
  P[idx] = (i == j ? 1.0f : 0.0f) + h * A[idx] + 0.5f * h * h * T2[idx]
         + (h * h * h * (1.0f / 6.0f)) * T3[idx];
}

__global__ void __launch_bounds__(256)
build_Q0(float* __restrict__ Q, const float* __restrict__ By, const float* __restrict__ dtp)
{
  const int idx = blockIdx.x * 256 + threadIdx.x;
  Q[idx] = (*dtp) * By[idx];
}

// ---------------------------------------------------------------------------
// Scan kernel: 8 workgroups x 256 threads (8 wave32s). Workgroup g owns batch
// rows [16g, 16g+16). Per step: x_next = x@P + y@Q (fp32 WMMA 16x16x4),
// u = x@Cu + y@Duy. P/Q B-fragments are loop-invariant per wave and live in
// 128 VGPRs; the y tile is software-pipelined (register prefetch + double-
// buffered LDS) so HBM latency is hidden behind the step's WMMA work.
// The u-GEMM duty goes to waves {0,3,5,6}: this set places exactly one heavy
// and one light wave on each SIMD under BOTH plausible wave->SIMD pairings
// ((2i,2i+1) and (i,i+4)), keeping the per-SIMD matrix-pipe load balanced.
// 2 workgroup barriers per step.
// ---------------------------------------------------------------------------
__global__ void __launch_bounds__(256)
scan_kernel(const float* __restrict__ obs,
            const float* __restrict__ x0,
            const float* __restrict__ CuT,
            const float* __restrict__ DuyT,
            const float* __restrict__ log_stds,
            const float* __restrict__ Pm,
            const float* __restrict__ Qm,
            float* __restrict__ outputs,
            float* __restrict__ x_final)
{
  extern __shared__ float sm[];
  float* sCu  = sm;                 // 128 x 68
  float* sDuy = sCu  + 128 * 68;    // 128 x 68
  float* sX   = sDuy + 128 * 68;    // 16 x 132 (state tile)
  float* sY   = sX   + 16 * 132;    // 2 x (16 x 132) double-buffered obs tile

  const int tid  = threadIdx.x;
  const int wv   = tid >> 5;        // wave 0..7
  const int lane = tid & 31;
  const int hi   = (lane >> 4) & 1;
  const int lr   = lane & 15;
  const int koff = hi << 1;
  const int b0   = blockIdx.x * 16;
  const int n0   = wv << 4;         // this wave's 16-column state tile

  // role assignment balanced under both wave->SIMD pairings
  const int heavy = (wv == 0) | (wv == 3) | (wv == 5) | (wv == 6);
  const int uidx  = (wv == 0) ? 0 : (wv == 3) ? 1 : (wv == 5) ? 2 : 3;  // heavy waves
  const int lidx  = (wv == 1) ? 0 : (wv == 2) ? 1 : (wv == 4) ? 2 : 3;  // light waves
  const int nu    = uidx << 4;              // u column tile for heavy waves
  const int lg    = (lidx << 5) + lane;     // 0..127 across the 4 light waves
  const int lrow  = lg >> 3;                // log-std row 0..15
  const int lj8   = (lg & 7) << 3;          // log-std col offset 0..56

  // ---- stage Cu, Duy (stride 68) ----
  for (int idx = tid; idx < 128 * 16; idx += 256) {
    const int r = idx >> 4, c4 = (idx & 15) << 2;
    *(float4*)(sCu  + r * 68 + c4) = *(const float4*)(CuT  + r * 64 + c4);
    *(float4*)(sDuy + r * 68 + c4) = *(const float4*)(DuyT + r * 64 + c4);
  }
  // ---- stage initial state rows ----
  for (int idx = tid; idx < 16 * 32; idx += 256) {
    const int r = idx >> 5, c4 = (idx & 31) << 2;
    *(float4*)(sX + r * 132 + c4) = *(const float4*)(x0 + (long long)(b0 + r) * 128 + c4);
  }

  // ---- hoist loop-invariant P / Q B-fragments into registers (128 VGPRs) ----
  v2f pB[32], qB[32];
#pragma unroll
  for (int kk = 0; kk < 32; ++kk) {
    const int k = (kk << 2) + koff;
    pB[kk].x = Pm[k * 128 + n0 + lr];
    pB[kk].y = Pm[(k + 1) * 128 + n0 + lr];
    qB[kk].x = Qm[k * 128 + n0 + lr];
    qB[kk].y = Qm[(k + 1) * 128 + n0 + lr];
  }

  // ---- preload log-std values (light waves store them each step) ----
  float ls[8];
#pragma unroll
  for (int j = 0; j < 8; ++j) ls[j] = log_stds[lj8 + j];

  const long long rowStride = (long long)Tn * 128;
  const int yr = tid >> 4, ycs = (tid & 15) << 3;   // per-thread y-load slice

  // ---- prefetch y(t=0) into registers ----
  float4 pf0, pf1;
  {
    const float* gp = obs + ((long long)(b0 + yr) * Tn) * 128 + ycs;
    pf0 = *(const float4*)gp;
    pf1 = *(const float4*)(gp + 4);
  }
  __syncthreads();

  for (int t = 0; t < Tn; ++t) {
    float* sYc = sY + (t & 1) * (16 * 132);
    // drain prefetch into this step's y buffer
    *(float4*)(sYc + yr * 132 + ycs)     = pf0;
    *(float4*)(sYc + yr * 132 + ycs + 4) = pf1;
    __syncthreads();   // sY cur ready; sX from previous step ready

    // issue next step's y loads; consumed only at the top of step t+1
    if (t + 1 < Tn) {
      const float* gp = obs + ((long long)(b0 + yr) * Tn + (t + 1)) * 128 + ycs;
      pf0 = *(const float4*)gp;
      pf1 = *(const float4*)(gp + 4);
    }

    // c = y @ Q  (wave tile: rows 0..15, cols n0..n0+15, K = OBS = 128)
    v8f c0 = V8F_ZERO, c1 = V8F_ZERO;
#pragma unroll
    for (int kk = 0; kk < 16; ++kk) {
      v2f a = *(const v2f*)(sYc + lr * 132 + (kk << 2) + koff);
      c0 = WMMA_F32X4(a, qB[kk], c0);
    }
#pragma unroll
    for (int kk = 16; kk < 32; ++kk) {
      v2f a = *(const v2f*)(sYc + lr * 132 + (kk << 2) + koff);
      c1 = WMMA_F32X4(a, qB[kk], c1);
    }

    if (heavy) {
      // u = x @ Cu + y @ Duy  (cols nu..nu+15 of OUT=64), store to outputs
      v8f u0 = V8F_ZERO, u1 = V8F_ZERO;
#pragma unroll
      for (int kk = 0; kk < 16; ++kk) {
        const int k = (kk << 2) + koff;
        v2f a = *(const v2f*)(sX + lr * 132 + k);
        v2f b;
        b.x = sCu[k * 68 + nu + lr];
        b.y = sCu[(k + 1) * 68 + nu + lr];
        u0 = WMMA_F32X4(a, b, u0);
      }
#pragma unroll
      for (int kk = 16; kk < 32; ++kk) {
        const int k = (kk << 2) + koff;
        v2f a = *(const v2f*)(sX + lr * 132 + k);
        v2f b;
        b.x = sCu[k * 68 + nu + lr];
        b.y = sCu[(k + 1) * 68 + nu + lr];
        u1 = WMMA_F32X4(a, b, u1);
      }
#pragma unroll
      for (int kk = 0; kk < 16; ++kk) {
        const int k = (kk << 2) + koff;
        v2f a = *(const v2f*)(sYc + lr * 132 + k);
        v2f b;
        b.x = sDuy[k * 68 + nu + lr];
        b.y = sDuy[(k + 1) * 68 + nu + lr];
        u0 = WMMA_F32X4(a, b, u0);
      }
#pragma unroll
      for (int kk = 16; kk < 32; ++kk) {
        const int k = (kk << 2) + koff;
        v2f a = *(const v2f*)(sYc + lr * 132 + k);
        v2f b;
        b.x = sDuy[k * 68 + nu + lr];
        b.y = sDuy[(k + 1) * 68 + nu + lr];
        u1 = WMMA_F32X4(a, b, u1);
      }
#pragma unroll
      for (int i = 0; i < 8; ++i) {
        outputs[(long long)(b0 + i + hi * 8) * rowStride + (long long)t * 128 + nu + lr] =
            u0[i] + u1[i];
      }
    } else {
      // constant log-std half of the output row
      float* op = outputs + (long long)(b0 + lrow) * rowStride + (long long)t * 128 + 64 + lj8;
      *(float4*)op       = make_float4(ls[0], ls[1], ls[2], ls[3]);
      *(float4*)(op + 4) = make_float4(ls[4], ls[5], ls[6], ls[7]);
    }

    // c += x @ P   (K = S = 128) -> c is x_next tile
#pragma unroll
    for (int kk = 0; kk < 16; ++kk) {
      v2f a = *(const v2f*)(sX + lr * 132 + (kk << 2) + koff);
      c0 = WMMA_F32X4(a, pB[kk], c0);
    }
#pragma unroll
    for (int kk = 16; kk < 32; ++kk) {
      v2f a = *(const v2f*)(sX + lr * 132 + (kk << 2) + koff);
      c1 = WMMA_F32X4(a, pB[kk], c1);
    }
    __syncthreads();   // every wave finished reading sX / sYc
#pragma unroll
    for (int i = 0; i < 8; ++i)
      sX[(i + hi * 8) * 132 + n0 + lr] = c0[i] + c1[i];
  }
  __syncthreads();

  // x_final
  for (int idx = tid; idx < 16 * 32; idx += 256) {
    const int r = idx >> 5, c4 = (idx & 31) << 2;
    *(float4*)(x_final + (long long)(b0 + r) * 128 + c4) = *(const float4*)(sX + r * 132 + c4);
  }
}

// ---------------------------------------------------------------------------
// Value MLP: rows = B*T. 2048 workgroups x 128 rows. fp32 WMMA for both
// Linear layers, VALU+shfl reduce for the final width-1 layer.
// ---------------------------------------------------------------------------
__global__ void __launch_bounds__(256)
mlp_kernel(const float* __restrict__ obs,
           const float* __restrict__ W0, const float* __restrict__ b0v,
           const float* __restrict__ W1, const float* __restrict__ b1v,
           const float* __restrict__ W2, const float* __restrict__ b2v,
           float* __restrict__ cur_value)
{
  extern __shared__ float sm[];
  float* sW0 = sm;                 // 128 x 68  (W0^T: [k=OBS][n=H])
  float* sW1 = sW0 + 128 * 68;     // 64 x 68   (W1^T)
  float* sO  = sW1 + 64 * 68;      // 128 x 132 obs tile
  float* sH  = sO  + 128 * 132;    // 128 x 68  hidden tile
  float* sB0 = sH  + 128 * 68;     // 64
  float* sB1 = sB0 + 64;           // 64
  float* sW2 = sB1 + 64;           // 64

  const int tid  = threadIdx.x;
  const int wv   = tid >> 5;
  const int lane = tid & 31;
  const int hi   = (lane >> 4) & 1;
  const int lr   = lane & 15;
  const int koff = hi << 1;
  const long long r0 = (long long)blockIdx.x * 128;
  const int rbase = wv << 4;       // this wave's 16-row block

  // stage transposed weights + biases
  for (int idx = tid; idx < 64 * 128; idx += 256) {
    const int n = idx >> 7, k = idx & 127;
    sW0[k * 68 + n] = W0[n * 128 + k];
  }
  for (int idx = tid; idx < 64 * 64; idx += 256) {
    const int n = idx >> 6, k = idx & 63;
    sW1[k * 68 + n] = W1[n * 64 + k];
  }
  if (tid < 64) { sB0[tid] = b0v[tid]; sB1[tid] = b1v[tid]; sW2[tid] = W2[tid]; }
  // stage obs tile
  for (int idx = tid; idx < 128 * 32; idx += 256) {
    const int r = idx >> 5, c4 = (idx & 31) << 2;
    *(float4*)(sO + r * 132 + c4) = *(const float4*)(obs + (r0 + r) * 128 + c4);
  }
  __syncthreads();

  // layer 1: h0 = tanh(obs @ W0^T + b0); 4 column tiles of H=64, K=128
#pragma unroll
  for (int ct = 0; ct < 4; ++ct) {
    const int n0 = ct << 4;
    v8f c = V8F_ZERO;
#pragma unroll
    for (int k = 0; k < 128; k += 4) {
      v2f a = *(const v2f*)(sO + (rbase + lr) * 132 + k + koff);
      v2f b;
      b.x = sW0[(k + koff) * 68 + n0 + lr];
      b.y = sW0[(k + koff + 1) * 68 + n0 + lr];
      c = WMMA_F32X4(a, b, c);
    }
    const float bb = sB0[n0 + lr];
#pragma unroll
    for (int i = 0; i < 8; ++i)
      sH[(rbase + i + hi * 8) * 68 + n0 + lr] = tanhf(c[i] + bb);
  }
  __syncthreads();

  // layer 2: h1 = tanh(h0 @ W1^T + b1); K=64; accumulate all tiles, then overwrite sH
  v8f c2[4];
#pragma unroll
  for (int ct = 0; ct < 4; ++ct) {
    const int n0 = ct << 4;
    v8f c = V8F_ZERO;
#pragma unroll
    for (int k = 0; k < 64; k += 4) {
      v2f a = *(const v2f*)(sH + (rbase + lr) * 68 + k + koff);
      v2f b;
      b.x = sW1[(k + koff) * 68 + n0 + lr];
      b.y = sW1[(k + koff + 1) * 68 + n0 + lr];
      c = WMMA_F32X4(a, b, c);
    }
    c2[ct] = c;
  }
  __syncthreads();
#pragma unroll
  for (int ct = 0; ct < 4; ++ct) {
    const int n0 = ct << 4;
    const float bb = sB1[n0 + lr];
#pragma unroll
    for (int i = 0; i < 8; ++i)
      sH[(rbase + i + hi * 8) * 68 + n0 + lr] = tanhf(c2[ct][i] + bb);
  }
  __syncthreads();

  // layer 3: v = h1 @ W2^T + b2  (2 lanes per row, shfl-combine)
  {
    float s = 0.0f;
#pragma unroll
    for (int j = 0; j < 32; ++j) {
      const int jj = hi * 32 + j;
      s = fmaf(sH[(rbase + lr) * 68 + jj], sW2[jj], s);
    }
    s += __shfl_xor(s, 16, 32);
    if (!hi) cur_value[r0 + rbase + lr] = s + b2v[0];
  }
}

// ---------------------------------------------------------------------------
extern "C" void kernel_launch(void* const* d_in, const int* in_sizes, int n_in,
                              void* d_out, int out_size, void* d_ws, size_t ws_size,
                              hipStream_t stream)
{
  const float* obs      = (const float*)d_in[0];
  const float* x0       = (const float*)d_in[1];
  const float* A_T      = (const float*)d_in[2];
  const float* By_T     = (const float*)d_in[3];
  const float* Cu_T     = (const float*)d_in[4];
  const float* Duy_T    = (const float*)d_in[5];
  const float* log_stds = (const float*)d_in[6];
  const float* W0       = (const float*)d_in[7];
  const float* b0       = (const float*)d_in[8];
  const float* W1       = (const float*)d_in[9];
  const float* b1       = (const float*)d_in[10];
  const float* W2       = (const float*)d_in[11];
  const float* b2       = (const float*)d_in[12];
  const float* dtp      = (const float*)d_in[13];

  float* ws = (float*)d_ws;
  float* T2 = ws;               // 128*128
  float* T3 = ws + 16384;
  float* Pm = ws + 32768;
  float* Qm = ws + 49152;       // total 256 KB of scratch

  float* outputs   = (float*)d_out;
  float* cur_value = outputs + OUT_ELEMS;     // B*T values
  float* x_final   = cur_value + ROWS;        // B*S values

  // --- one-time transition-matrix precompute (serialized on stream) ---
  gemm128_acc<<<64, 256, 0, stream>>>(T2, A_T, A_T, dtp, 0, 0);   // T2 = A^2
  gemm128_acc<<<64, 256, 0, stream>>>(T3, T2, A_T, dtp, 0, 0);    // T3 = A^3
  build_P   <<<64, 256, 0, stream>>>(Pm, A_T, T2, T3, dtp);       // P  = I+hA+..A^3
  gemm128_acc<<<64, 256, 0, stream>>>(Pm, T2, T2, dtp, 3, 1);     // P += h^4/24 A^4
  build_Q0  <<<64, 256, 0, stream>>>(Qm, By_T, dtp);              // Q  = hB
  gemm128_acc<<<64, 256, 0, stream>>>(Qm, By_T, A_T, dtp, 1, 1);  // Q += h^2/2 BA
  gemm128_acc<<<64, 256, 0, stream>>>(Qm, By_T, T2, dtp, 2, 1);   // Q += h^3/6 BA^2
  gemm128_acc<<<64, 256, 0, stream>>>(Qm, By_T, T3, dtp, 3, 1);   // Q += h^4/24 BA^3

  // --- scan (8 workgroups, one per 16 batch rows) ---
  const size_t scan_smem = (size_t)(2 * 128 * 68 + 16 * 132 + 2 * 16 * 132) * sizeof(float);
  hipFuncSetAttribute(reinterpret_cast<const void*>(scan_kernel),
                      hipFuncAttributeMaxDynamicSharedMemorySize, (int)scan_smem);
  scan_kernel<<<Bn / 16, 256, scan_smem, stream>>>(obs, x0, Cu_T, Duy_T, log_stds,
                                                   Pm, Qm, outputs, x_final);

  // --- value MLP (2048 workgroups x 128 rows) ---
  const size_t mlp_smem = (size_t)(128 * 68 + 64 * 68 + 128 * 132 + 128 * 68 + 192) * sizeof(float);
  hipFuncSetAttribute(reinterpret_cast<const void*>(mlp_kernel),
                      hipFuncAttributeMaxDynamicSharedMemorySize, (int)mlp_smem);
  mlp_kernel<<<(int)(ROWS / 128), 256, mlp_smem, stream>>>(obs, W0, b0, W1, b1, W2, b2,
                                                           cur_value);
}